// SimpleGRU_74199855006382
// MI455X (gfx1250) — compile-verified
//
#include <hip/hip_runtime.h>

typedef float v2f __attribute__((ext_vector_type(2)));
typedef float v8f __attribute__((ext_vector_type(8)));

constexpr int SEQLEN  = 512;
constexpr int BATCH   = 64;
constexpr int EMBD    = 256;
constexpr int HID     = 512;
constexpr int NOUT    = 5;
constexpr int NWG     = 16;          // workgroups in persistent kernel
constexpr int HB      = HID / NWG;   // hidden columns owned per WG = 32
constexpr int NTHR    = 256;         // 8 waves of 32

__device__ __forceinline__ v8f wmma_f32(v2f a, v2f b, v8f c) {
  // D = A(16x4) * B(4x16) + C(16x16), all f32, wave32
  return __builtin_amdgcn_wmma_f32_16x16x4_f32(
      /*neg_a=*/false, a, /*neg_b=*/false, b,
      /*c_mod=*/(short)0, c, /*reuse_a=*/false, /*reuse_b=*/false);
}

__global__ void gru_init_kernel(float* __restrict__ h0, float* __restrict__ h1,
                                unsigned* __restrict__ bar) {
  int idx = blockIdx.x * blockDim.x + threadIdx.x;
  int stride = gridDim.x * blockDim.x;
  for (int q = idx; q < BATCH * HID; q += stride) { h0[q] = 0.f; h1[q] = 0.f; }
  if (idx == 0) *bar = 0u;
}

// Persistent recurrent kernel: NWG blocks, each owns HB hidden columns.
// Per step: stage h into LDS, WMMA the three gate projections (x-part from
// gathered embedding rows, h-part from LDS), fuse the GRU pointwise update
// in-register, write new h to the alternate buffer, grid-barrier.
__global__ void __launch_bounds__(NTHR)
gru_recurrent_kernel(const int* __restrict__ seq, const int* __restrict__ lengths,
                     const float* __restrict__ emb,
                     const float* __restrict__ w_ih, const float* __restrict__ w_hh,
                     const float* __restrict__ b_ih, const float* __restrict__ b_hh,
                     float* __restrict__ hbuf0, float* __restrict__ hbuf1,
                     unsigned* __restrict__ bar) {
  extern __shared__ char smem_raw[];
  float* hs  = (float*)smem_raw;                                // [BATCH][HID]
  int* sidx  = (int*)(smem_raw + (size_t)BATCH * HID * 4);      // [BATCH]
  int* cnt   = sidx + BATCH;                                    // [SEQLEN]

  const int tid  = threadIdx.x;
  const int wg   = blockIdx.x;
  const int lane = tid & 31;
  const int wave = tid >> 5;        // 0..7
  const int nl   = lane & 15;       // 0..15
  const int hi   = lane >> 4;       // 0/1 (K half for A/B fragments)
  const int ko   = hi * 2;          // K offset within a k0 group of 4

  // Precompute active-batch count per timestep (lengths sorted descending).
  for (int t = tid; t < SEQLEN; t += NTHR) {
    int c = 0;
    for (int b = 0; b < BATCH; ++b) c += (lengths[b] > t) ? 1 : 0;
    cnt[t] = c;
  }

  // Fixed wave -> tile mapping: i = batch row-tile (16 rows), cb = 16-col block.
  const int i  = wave >> 1;
  const int cb = wave & 1;
  const int colbase = wg * HB;
  const int cl = colbase + cb * 16 + nl;    // this lane's hidden column (B-frag N)

  // Per-gate weight rows for column cl (B matrix = W^T, contiguous in K).
  const float* wih_r = w_ih + (size_t)(0 * HID + cl) * EMBD;
  const float* wih_z = w_ih + (size_t)(1 * HID + cl) * EMBD;
  const float* wih_n = w_ih + (size_t)(2 * HID + cl) * EMBD;
  const float* whh_r = w_hh + (size_t)(0 * HID + cl) * HID;
  const float* whh_z = w_hh + (size_t)(1 * HID + cl) * HID;
  const float* whh_n = w_hh + (size_t)(2 * HID + cl) * HID;
  const float bias_r = b_ih[0 * HID + cl] + b_hh[0 * HID + cl];
  const float bias_z = b_ih[1 * HID + cl] + b_hh[1 * HID + cl];
  const float bxn    = b_ih[2 * HID + cl];
  const float bhn    = b_hh[2 * HID + cl];

  unsigned gen = 0;
  for (int t = 0; t < SEQLEN; ++t) {
    float* hsrc = (t & 1) ? hbuf1 : hbuf0;
    float* hdst = (t & 1) ? hbuf0 : hbuf1;

    // Stage h panel (full H needed as GEMM K-dim) and this step's token ids.
    {
      float4* hs4 = (float4*)hs;
      const float4* src4 = (const float4*)hsrc;
      for (int q = tid; q < BATCH * HID / 4; q += NTHR) hs4[q] = src4[q];
      if (tid < BATCH) sidx[tid] = seq[t * BATCH + tid];
    }
    __syncthreads();

    const int Bt = cnt[t];
    const int mt = (Bt + 15) >> 4;      // active 16-row tiles

    if (i < mt) {
      const int m = i * 16 + nl;                        // A-frag row (batch)
      const float* arow_x = emb + (size_t)sidx[m] * EMBD;
      const float* arow_h = hs + (size_t)m * HID;

      v8f acc_r = {}; v8f acc_z = {}; v8f acc_xn = {}; v8f acc_hn = {};

      // Input projection: K = 0..EMBD over gathered embedding rows.
      #pragma unroll 4
      for (int k0 = 0; k0 < EMBD; k0 += 4) {
        v2f a = *(const v2f*)(arow_x + k0 + ko);
        acc_r  = wmma_f32(a, *(const v2f*)(wih_r + k0 + ko), acc_r);
        acc_z  = wmma_f32(a, *(const v2f*)(wih_z + k0 + ko), acc_z);
        acc_xn = wmma_f32(a, *(const v2f*)(wih_n + k0 + ko), acc_xn);
      }
      // Hidden projection: K = 0..HID over LDS-staged h.
      #pragma unroll 4
      for (int k0 = 0; k0 < HID; k0 += 4) {
        v2f a = *(const v2f*)(arow_h + k0 + ko);
        acc_r  = wmma_f32(a, *(const v2f*)(whh_r + k0 + ko), acc_r);
        acc_z  = wmma_f32(a, *(const v2f*)(whh_z + k0 + ko), acc_z);
        acc_hn = wmma_f32(a, *(const v2f*)(whh_n + k0 + ko), acc_hn);
      }

      // Fused GRU pointwise update; C/D layout: vgpr v -> row i*16+v+hi*8, col nl.
      #pragma unroll
      for (int v = 0; v < 8; ++v) {
        int b = i * 16 + v + hi * 8;
        float r = 1.f / (1.f + expf(-(acc_r[v] + bias_r)));
        float z = 1.f / (1.f + expf(-(acc_z[v] + bias_z)));
        float n = tanhf(acc_xn[v] + bxn + r * (acc_hn[v] + bhn));
        float hold = hs[(size_t)b * HID + cl];
        float hnew = (1.f - z) * n + z * hold;
        hdst[(size_t)b * HID + cl] = (b < Bt) ? hnew : hold;   // packed-seq freeze
      }
    }

    // Copy fully-frozen rows (>= mt*16) forward in this WG's columns.
    {
      int fr0 = mt * 16;
      int nrows = BATCH - fr0;
      for (int q = tid; q < nrows * HB; q += NTHR) {
        int b = fr0 + q / HB;
        int c = colbase + (q % HB);
        hdst[(size_t)b * HID + c] = hs[(size_t)b * HID + c];
      }
    }

    // One grid barrier per step (arrive + spin on monotonic counter).
    __syncthreads();
    if (tid == 0) {
      __threadfence();
      ++gen;
      __hip_atomic_fetch_add(bar, 1u, __ATOMIC_ACQ_REL, __HIP_MEMORY_SCOPE_AGENT);
      unsigned target = gen * (unsigned)NWG;
      while (__hip_atomic_load(bar, __ATOMIC_ACQUIRE, __HIP_MEMORY_SCOPE_AGENT) < target)
        __builtin_amdgcn_s_sleep(1);
      __threadfence();
    }
    __syncthreads();
  }
}

// Output head: logits = h_last @ w_out^T + b_out, then log_softmax per row.
__global__ void __launch_bounds__(320)
gru_out_kernel(const float* __restrict__ hfin, const float* __restrict__ w_out,
               const float* __restrict__ b_out, float* __restrict__ out) {
  __shared__ float logits[BATCH][NOUT];
  int tid = threadIdx.x;
  if (tid < BATCH * NOUT) {
    int b = tid / NOUT, o = tid % NOUT;
    const float* hr = hfin + (size_t)b * HID;
    const float* wr = w_out + (size_t)o * HID;
    float acc = b_out[o];
    for (int k = 0; k < HID; ++k) acc = fmaf(hr[k], wr[k], acc);
    logits[b][o] = acc;
  }
  __syncthreads();
  if (tid < BATCH) {
    float m = logits[tid][0];
    #pragma unroll
    for (int o = 1; o < NOUT; ++o) m = fmaxf(m, logits[tid][o]);
    float s = 0.f;
    #pragma unroll
    for (int o = 0; o < NOUT; ++o) s += expf(logits[tid][o] - m);
    float ls = logf(s);
    #pragma unroll
    for (int o = 0; o < NOUT; ++o) out[tid * NOUT + o] = logits[tid][o] - m - ls;
  }
}

extern "C" void kernel_launch(void* const* d_in, const int* in_sizes, int n_in,
                              void* d_out, int out_size, void* d_ws, size_t ws_size,
                              hipStream_t stream) {
  (void)in_sizes; (void)n_in; (void)out_size; (void)ws_size;
  const int*   seq     = (const int*)d_in[0];
  const int*   lengths = (const int*)d_in[1];
  const float* emb     = (const float*)d_in[2];
  const float* w_ih    = (const float*)d_in[3];
  const float* w_hh    = (const float*)d_in[4];
  const float* b_ih    = (const float*)d_in[5];
  const float* b_hh    = (const float*)d_in[6];
  const float* w_out   = (const float*)d_in[7];
  const float* b_out   = (const float*)d_in[8];
  float* out = (float*)d_out;

  char* ws = (char*)d_ws;
  float* h0 = (float*)ws;                                     // 128 KB
  float* h1 = (float*)(ws + (size_t)BATCH * HID * 4);         // 128 KB
  unsigned* bar = (unsigned*)(ws + 2 * (size_t)BATCH * HID * 4);

  gru_init_kernel<<<64, 256, 0, stream>>>(h0, h1, bar);

  size_t lds = (size_t)BATCH * HID * 4 + BATCH * 4 + SEQLEN * 4;  // ~130.3 KB
  gru_recurrent_kernel<<<NWG, NTHR, lds, stream>>>(
      seq, lengths, emb, w_ih, w_hh, b_ih, b_hh, h0, h1, bar);

  // 512 steps end with dst = h0.
  gru_out_kernel<<<1, 320, 0, stream>>>(h0, w_out, b_out, out);
}